// yoloLoss_53884659695988
// MI455X (gfx1250) — compile-verified
//
#include <hip/hip_runtime.h>
#include <math.h>

#define SS 14
#define NCLS 20
#define STEPF (1.0f/14.0f)
#define EPSF 1e-7f
#define NBATCH 4096
#define CELLS (NBATCH*SS*SS)          /* 802816 */
#define THREADS 256
#define BLOCKS 392
#define CHUNK 256                     /* cells per chunk (== THREADS)      */
#define CHUNK_FLOATS (CHUNK*30)       /* 7680 floats per tensor per chunk  */
#define CHUNK_F4 (CHUNK_FLOATS/4)     /* 1920 float4 per tensor per chunk  */
#define CHUNK_BYTES (CHUNK_FLOATS*4)  /* 30720 bytes                        */
#define CPB 8                         /* chunks per block: 392*8*256=CELLS */
#define NQ 6
#define REC 16                        /* floats per block partial record   */

typedef __attribute__((ext_vector_type(2))) float v2f;
typedef __attribute__((ext_vector_type(8))) float v8f;

// ---------------------------------------------------------------------------
// Issue one chunk's async global->LDS copy: 2 tensors x 1920 float4 slots,
// 15 slots/thread, fully coalesced b128 transfers tracked by ASYNCcnt.
// Slot->tensor boundary (1920) is a multiple of 32*? -> falls at tid==128,
// so addressing is uniform per wave.
// ---------------------------------------------------------------------------
__device__ __forceinline__ void async_issue_chunk(uint64_t pbase, uint64_t tbase,
                                                  const float* dstP,
                                                  const float* dstT, int tid) {
#pragma unroll
  for (int k = 0; k < 15; ++k) {
    const int idx  = tid + THREADS * k;            // 0..3839 float4 slots
    const bool isP = idx < CHUNK_F4;
    const int rel  = isP ? idx : (idx - CHUNK_F4);
    const uint32_t lds = (uint32_t)(uintptr_t)(isP ? dstP : dstT)
                       + (uint32_t)rel * 16u;      // low 32 bits == LDS offset
    const uint64_t ga  = (isP ? pbase : tbase) + (uint64_t)rel * 16u;
    asm volatile("global_load_async_to_lds_b128 %0, %1, off"
                 :: "v"(lds), "v"(ga) : "memory");
  }
}

// ---------------------------------------------------------------------------
// Kernel 1: double-buffered async LDS staging + per-cell loss partials.
// Partials per block: [obj, noobj, bbox_diou, n_obj, focal, n_sig, 0..0]
// ---------------------------------------------------------------------------
__global__ __launch_bounds__(THREADS)
void yolo_partial(const float* __restrict__ pred,
                  const float* __restrict__ target,
                  float* __restrict__ partials) {
  __shared__ alignas(16) float smP[2][CHUNK_FLOATS];   // 2 x 30720 B
  __shared__ alignas(16) float smT[2][CHUNK_FLOATS];
  __shared__ float red[NQ][THREADS];

  float a_obj = 0.f, a_noobj = 0.f, a_bbox = 0.f, a_nobj = 0.f,
        a_focal = 0.f, a_nsig = 0.f;
  const int tid = threadIdx.x;
  const int ch0 = blockIdx.x * CPB;                    // first chunk index
  const uint64_t pb = (uint64_t)(uintptr_t)pred;
  const uint64_t tb = (uint64_t)(uintptr_t)target;

  // prime buffer 0 with chunk ch0
  async_issue_chunk(pb + (uint64_t)ch0 * CHUNK_BYTES,
                    tb + (uint64_t)ch0 * CHUNK_BYTES, smP[0], smT[0], tid);

  for (int k = 0; k < CPB; ++k) {
    const int buf = k & 1;
    if (k + 1 < CPB) {
      const uint64_t off = (uint64_t)(ch0 + k + 1) * CHUNK_BYTES;
      async_issue_chunk(pb + off, tb + off, smP[buf ^ 1], smT[buf ^ 1], tid);
      asm volatile("s_wait_asynccnt 0xf" ::: "memory");  // chunk k complete
    } else {
      asm volatile("s_wait_asynccnt 0x0" ::: "memory");
    }
    __syncthreads();                                     // whole chunk visible

    // ---- compute this thread's cell from LDS ----
    const int cell = (ch0 + k) * CHUNK + tid;
    float p[30], t[30];
    const float2* lp = (const float2*)&smP[buf][tid * 30];
    const float2* lt = (const float2*)&smT[buf][tid * 30];
#pragma unroll
    for (int w = 0; w < 15; ++w) {
      float2 a = lp[w]; float2 b = lt[w];
      p[2*w] = a.x; p[2*w+1] = a.y;
      t[2*w] = b.x; t[2*w+1] = b.y;
    }

    const int i = cell % SS;
    const int j = (cell / SS) % SS;
    const float fi = (float)i, fj = (float)j;

    float pcx[2], pcy[2], pw[2], ph[2], pcf[2];
    float tcx[2], tcy[2], tw[2], th_[2], tcf[2];
    float iou_c[2];
#pragma unroll
    for (int b = 0; b < 2; ++b) {
      pcx[b]=p[b*5+0]; pcy[b]=p[b*5+1]; pw[b]=p[b*5+2]; ph[b]=p[b*5+3]; pcf[b]=p[b*5+4];
      tcx[b]=t[b*5+0]; tcy[b]=t[b*5+1]; tw[b]=t[b*5+2]; th_[b]=t[b*5+3]; tcf[b]=t[b*5+4];
      // _convert_cell_box (corner form), clip(out, 0)
      float px_ = fmaxf((pcx[b]+fi)*STEPF - pw[b]*0.5f, 0.f);
      float py_ = fmaxf((pcy[b]+fj)*STEPF - ph[b]*0.5f, 0.f);
      float pw_ = fmaxf(pw[b], 0.f);
      float ph2 = fmaxf(ph[b], 0.f);
      float tx_ = fmaxf((tcx[b]+fi)*STEPF - tw[b]*0.5f, 0.f);
      float ty_ = fmaxf((tcy[b]+fj)*STEPF - th_[b]*0.5f, 0.f);
      float tw_ = fmaxf(tw[b], 0.f);
      float th2 = fmaxf(th_[b], 0.f);
      // _paired_iou
      float iw = pw_ + tw_ - (fmaxf(px_+pw_, tx_+tw_) - fminf(px_, tx_));
      float ih = ph2 + th2 - (fmaxf(py_+ph2, ty_+th2) - fminf(py_, ty_));
      float inter = fmaxf(iw, 0.f) * fmaxf(ih, 0.f);
      float uni = pw_*ph2 + tw_*th2 - inter + EPSF;
      iou_c[b] = inter / uni;
    }
    const int  maxi = (iou_c[1] > iou_c[0]) ? 1 : 0;    // argmax, tie -> 0
    const bool sig  = tcf[1] > 0.f;

#pragma unroll
    for (int b = 0; b < 2; ++b) {
      const bool keep = (!sig) || (b == maxi);
      const bool om   = (tcf[b] > 0.f) && keep;
      const float fm  = om ? 1.f : 0.f;
      float dc = pcf[b] - tcf[b]; dc = dc * dc;
      a_obj   += fm * dc;
      a_noobj += (1.f - fm) * dc;
      // DIoU on raw (un-converted) boxes
      float px1 = pcx[b]-pw[b]*0.5f,  px2 = pcx[b]+pw[b]*0.5f;
      float py1 = pcy[b]-ph[b]*0.5f,  py2 = pcy[b]+ph[b]*0.5f;
      float tx1 = tcx[b]-tw[b]*0.5f,  tx2 = tcx[b]+tw[b]*0.5f;
      float ty1 = tcy[b]-th_[b]*0.5f, ty2 = tcy[b]+th_[b]*0.5f;
      float inter = fmaxf(fminf(px2,tx2)-fmaxf(px1,tx1), 0.f)
                  * fmaxf(fminf(py2,ty2)-fmaxf(py1,ty1), 0.f);
      float a1 = fmaxf(px2-px1, 0.f) * fmaxf(py2-py1, 0.f);
      float a2 = fmaxf(tx2-tx1, 0.f) * fmaxf(ty2-ty1, 0.f);
      float iou = fmaxf(inter / (a1 + a2 - inter + EPSF), EPSF);
      float dx = pcx[b]-tcx[b], dy = pcy[b]-tcy[b];
      float cd = dx*dx + dy*dy;
      float ddx = fmaxf(px2,tx2) - fminf(px1,tx1);
      float ddy = fmaxf(py2,ty2) - fminf(py1,ty1);
      float diag = ddx*ddx + ddy*ddy + EPSF;
      a_bbox += fm * (1.f - sqrtf(iou) + cd / diag);
      a_nobj += fm;
    }

    const float sm = sig ? 1.f : 0.f;
    float fsum = 0.f;
#pragma unroll
    for (int q = 0; q < NCLS; ++q) {
      float d = p[10+q] - t[10+q];
      float ce = d * d;
      float pt = fminf(fmaxf(expf(-ce), EPSF), 1.f);
      float ompt = 1.f - pt;
      fsum += ompt * ompt * ce;
    }
    a_focal += sm * fsum;
    a_nsig  += sm;

    __syncthreads();   // all reads of this buffer done before it is reused
  }

  // ---- block reduction ----
  red[0][tid]=a_obj; red[1][tid]=a_noobj; red[2][tid]=a_bbox;
  red[3][tid]=a_nobj; red[4][tid]=a_focal; red[5][tid]=a_nsig;
  __syncthreads();
  for (int s = THREADS/2; s > 0; s >>= 1) {
    if (tid < s) {
#pragma unroll
      for (int q = 0; q < NQ; ++q) red[q][tid] += red[q][tid+s];
    }
    __syncthreads();
  }
  if (tid == 0) {
    float* out = partials + (size_t)blockIdx.x * REC;
#pragma unroll
    for (int q = 0; q < REC; ++q) out[q] = (q < NQ) ? red[q][0] : 0.f;
  }
}

// ---------------------------------------------------------------------------
// Kernel 2: single wave32. WMMA ones-trick reduction:
//   D = ones(16x4) x B(4x16) + C  =>  D[m,n] += sum_k B[k,n]
// Column n carries quantity n; B column n lives in lanes {n, n+16} x {v0,v1},
// so each iteration folds 4 records regardless of exact K->slot order.
// ---------------------------------------------------------------------------
__global__ __launch_bounds__(32)
void yolo_finalize(const float* __restrict__ partials, float* __restrict__ out) {
  const int lane = threadIdx.x;
  const int n    = lane & 15;
  const int half = lane >> 4;
  v8f acc = {};
  v2f ones; ones.x = 1.0f; ones.y = 1.0f;
  for (int r = 0; r < BLOCKS; r += 4) {       // BLOCKS % 4 == 0
    const int r0 = r + 2 * half;
    v2f b;
    b.x = partials[(size_t)r0 * REC + n];
    b.y = partials[(size_t)(r0 + 1) * REC + n];
    acc = __builtin_amdgcn_wmma_f32_16x16x4_f32(
        false, ones, false, b, (short)0, acc, false, false);
  }
  const float c0 = acc[0];                    // D[0][n] on lanes 0..15
  const float s_obj  = __shfl(c0, 0, 32);
  const float s_no   = __shfl(c0, 1, 32);
  const float s_bbox = __shfl(c0, 2, 32);
  const float s_nobj = __shfl(c0, 3, 32);
  const float s_foc  = __shfl(c0, 4, 32);
  const float s_nsig = __shfl(c0, 5, 32);
  if (lane == 0) {
    const float bbox_loss = (s_nobj > 0.f) ? (s_bbox / fmaxf(s_nobj, 1.f)) : 0.f;
    const float ncls      = s_nsig * (float)NCLS;
    const float cls_loss  = (ncls > 0.f) ? (s_foc / fmaxf(ncls, 1.f)) : 0.f;
    const float total = s_obj + 0.5f * s_no + 5.0f * bbox_loss + cls_loss;
    out[0] = total / (float)NBATCH;
  }
}

extern "C" void kernel_launch(void* const* d_in, const int* in_sizes, int n_in,
                              void* d_out, int out_size, void* d_ws, size_t ws_size,
                              hipStream_t stream) {
  const float* pred   = (const float*)d_in[0];
  const float* target = (const float*)d_in[1];
  float* partials = (float*)d_ws;             // 392*16*4 = 25 KB
  yolo_partial<<<BLOCKS, THREADS, 0, stream>>>(pred, target, partials);
  yolo_finalize<<<1, 32, 0, stream>>>(partials, (float*)d_out);
}